// MultiPoolingModule2_60198261621161
// MI455X (gfx1250) — compile-verified
//
#include <hip/hip_runtime.h>
#include <hip/hip_bf16.h>
#include <stdint.h>

constexpr int FDIM   = 256;   // feature dim
constexpr int NPGc   = 512;   // nodes per group
constexpr int KSEL   = 154;   // ceil(0.3 * 512)
constexpr float BN_EPS = 1e-5f;
constexpr float LN_EPS = 1e-5f;

typedef float v2f __attribute__((ext_vector_type(2)));
typedef float v8f __attribute__((ext_vector_type(8)));

// ---------------------------------------------------------------- helpers
__device__ __forceinline__ float wave_sum32(float v) {
#pragma unroll
  for (int m = 16; m >= 1; m >>= 1) v += __shfl_xor(v, m, 32);
  return v;
}

// f32 WMMA: D(16x16) = A(16x4) * B(4x16) + C   (CDNA5 V_WMMA_F32_16X16X4_F32)
__device__ __forceinline__ v8f wmma_f32_k4(v2f a, v2f b, v8f c) {
  // 8 args: (neg_a, A, neg_b, B, c_mod, C, reuse_a, reuse_b)
  return __builtin_amdgcn_wmma_f32_16x16x4_f32(false, a, false, b, (short)0, c,
                                               false, false);
}

// ---------------------------------------------------------------- K1: h = x @ w + b, deg = 1
__global__ void k_gcn(const float* __restrict__ x, const float* __restrict__ w,
                      const float* __restrict__ b, float* __restrict__ h,
                      float* __restrict__ degf, int N) {
  __shared__ float sw[FDIM];
  int tid = threadIdx.x;
  for (int i = tid; i < FDIM; i += 256) sw[i] = w[i];
  __syncthreads();
  int wave = tid >> 5, lane = tid & 31;
  int row = blockIdx.x * 8 + wave;
  if (row >= N) return;
  const float* xr = x + (size_t)row * FDIM;
  float acc = 0.f;
#pragma unroll
  for (int c = 0; c < 2; ++c) {
    int off = lane * 4 + c * 128;
    float4 v = *(const float4*)(xr + off);
    acc += v.x * sw[off] + v.y * sw[off + 1] + v.z * sw[off + 2] + v.w * sw[off + 3];
  }
  acc = wave_sum32(acc);
  if (lane == 0) {
    h[row]    = acc + b[0];
    degf[row] = 1.0f;            // reference: deg = segment_sum(ones) + 1
  }
}

// ---------------------------------------------------------------- K2: degree count
__global__ void k_deg(const int* __restrict__ dst, float* __restrict__ degf, int E) {
  int e = blockIdx.x * blockDim.x + threadIdx.x;
  if (e < E) atomicAdd(&degf[dst[e]], 1.0f);
}

// ---------------------------------------------------------------- K3: dinv, score = h/deg
__global__ void k_dinv(const float* __restrict__ h, const float* __restrict__ degf,
                       float* __restrict__ dinv, float* __restrict__ score, int N) {
  int i = blockIdx.x * blockDim.x + threadIdx.x;
  if (i < N) {
    float di = rsqrtf(degf[i]);
    dinv[i]  = di;
    score[i] = h[i] * di * di;   // h / deg self-term
  }
}

// ---------------------------------------------------------------- K4: edge scatter-add
__global__ void k_edge(const int* __restrict__ src, const int* __restrict__ dst,
                       const float* __restrict__ h, const float* __restrict__ dinv,
                       float* __restrict__ score, int E) {
  int e = blockIdx.x * blockDim.x + threadIdx.x;
  if (e < E) {
    int s = src[e], d = dst[e];
    atomicAdd(&score[d], dinv[s] * dinv[d] * h[s]);
  }
}

// ---------------------------------------------------------------- K5: per-group top-K (bitonic)
// key = (orderedFloat(score) << 32) | (NPG-1 - idx)  -> descending sort picks
// largest scores, smallest index first on ties (matches jax.lax.top_k).
__global__ void k_topk(const float* __restrict__ score, int* __restrict__ topk) {
  __shared__ unsigned long long keys[NPGc];
  int g = blockIdx.x, tid = threadIdx.x;
  const float* s = score + (size_t)g * NPGc;
  for (int i = tid; i < NPGc; i += 256) {
    unsigned u = __float_as_uint(s[i]);
    u = (u & 0x80000000u) ? ~u : (u | 0x80000000u);
    keys[i] = ((unsigned long long)u << 32) | (unsigned)(NPGc - 1 - i);
  }
  __syncthreads();
  for (int k = 2; k <= NPGc; k <<= 1) {
    for (int j = k >> 1; j > 0; j >>= 1) {
      for (int t = tid; t < NPGc; t += 256) {
        int ixj = t ^ j;
        if (ixj > t) {
          bool descend = ((t & k) == 0);
          unsigned long long a = keys[t], bb = keys[ixj];
          if (descend ? (a < bb) : (a > bb)) { keys[t] = bb; keys[ixj] = a; }
        }
      }
      __syncthreads();
    }
  }
  if (tid < KSEL)
    topk[g * KSEL + tid] = (int)(NPGc - 1 - (unsigned)(keys[tid] & 0xFFFFFFFFu));
}

// ---------------------------------------------------------------- K6: gather + max/mean/sum pool
__global__ void k_pool(const float* __restrict__ x, const int* __restrict__ topk,
                       float* __restrict__ pooled) {
  int g = blockIdx.x, f = threadIdx.x;            // blockDim.x == FDIM
  __shared__ int idx[KSEL];
  for (int i = threadIdx.x; i < KSEL; i += blockDim.x) idx[i] = topk[g * KSEL + i];
  __syncthreads();
  float mx = -3.402823466e38f, sum = 0.f;
  for (int j = 0; j < KSEL; ++j) {
    size_t node = (size_t)g * NPGc + idx[j];
    float v = x[node * FDIM + f];
    if (j + 1 < KSEL)
      __builtin_prefetch(&x[((size_t)g * NPGc + idx[j + 1]) * FDIM + f], 0, 0);
    mx = fmaxf(mx, v);
    sum += v;
  }
  float* p = pooled + (size_t)g * 3 * FDIM;
  p[f]            = mx;
  p[FDIM + f]     = sum * (1.0f / KSEL);
  p[2 * FDIM + f] = sum;
}

// ---------------------------------------------------------------- K7: BN -> LN1 -> MLP (WMMA) -> +res -> LN2
// block = 128 threads (4 waves), one 16-row tile of the 768-row (G*3, F) matrix.
__global__ __launch_bounds__(128) void k_mlp(
    const float* __restrict__ pooled,
    const float* __restrict__ bng, const float* __restrict__ bnb,
    const float* __restrict__ bnm, const float* __restrict__ bnv,
    const float* __restrict__ ln1g, const float* __restrict__ ln1b,
    const float* __restrict__ w11, const float* __restrict__ b11,
    const float* __restrict__ w12, const float* __restrict__ b12,
    const float* __restrict__ ln2g, const float* __restrict__ ln2b,
    float* __restrict__ out) {
  __shared__ float sX1[16][FDIM + 1];   // BN output (residual input)
  __shared__ float sH1[16][FDIM + 1];   // LN1 output (GEMM1 A)
  __shared__ float sT1[16][64 + 1];     // relu(GEMM1) (GEMM2 A)
  __shared__ float sY [16][FDIM + 1];   // GEMM2 + bias + residual

  const int tid = threadIdx.x;
  const int R0  = blockIdx.x * 16;

  // ---- Phase A: BN + LN1 (8 threads per row, 32 features each)
  {
    int row = tid >> 3, sub = tid & 7;
    int grow = R0 + row;
    int ty = grow % 3;                       // pooled row type (max/mean/sum)
    const float* pr = pooled + (size_t)grow * FDIM;
    float s = 0.f, s2 = 0.f;
#pragma unroll 8
    for (int j = 0; j < 32; ++j) {
      int f = sub * 32 + j;
      float v  = pr[f];
      float xv = (v - bnm[ty * FDIM + f]) * rsqrtf(bnv[ty * FDIM + f] + BN_EPS) *
                     bng[ty * FDIM + f] + bnb[ty * FDIM + f];
      sX1[row][f] = xv;
      s += xv; s2 += xv * xv;
    }
#pragma unroll
    for (int m = 4; m >= 1; m >>= 1) { s += __shfl_xor(s, m, 32); s2 += __shfl_xor(s2, m, 32); }
    float mean = s * (1.0f / FDIM);
    float var  = s2 * (1.0f / FDIM) - mean * mean;
    float rstd = rsqrtf(var + LN_EPS);
#pragma unroll 8
    for (int j = 0; j < 32; ++j) {
      int f = sub * 32 + j;
      sH1[row][f] = (sX1[row][f] - mean) * rstd * ln1g[f] + ln1b[f];
    }
  }
  __syncthreads();

  const int lane = tid & 31;
  const int half = lane >> 4;     // K-half selector per WMMA layout
  const int lid  = lane & 15;     // M (A) / N (B,D) position
  const int wv   = tid >> 5;      // wave id 0..3

  // ---- Phase B: GEMM1  h1(16x256) @ w11(256x64) -> relu -> sT1(16x64)
  {
    int ncol0 = wv * 16;          // each wave owns one 16-col N tile
    v8f acc = {0.f, 0.f, 0.f, 0.f, 0.f, 0.f, 0.f, 0.f};
    for (int k = 0; k < FDIM; k += 4) {
      int kk = k + 2 * half;
      v2f a, b;
      a.x = sH1[lid][kk];     a.y = sH1[lid][kk + 1];
      b.x = w11[kk * 64 + ncol0 + lid];
      b.y = w11[(kk + 1) * 64 + ncol0 + lid];
      acc = wmma_f32_k4(a, b, acc);
    }
#pragma unroll
    for (int r = 0; r < 8; ++r) {
      int row = r + 8 * half, col = ncol0 + lid;
      sT1[row][col] = fmaxf(acc[r] + b11[col], 0.f);
    }
  }
  __syncthreads();

  // ---- Phase C: GEMM2  t1(16x64) @ w12(64x256) + b12 + x1 -> sY
  {
#pragma unroll
    for (int q = 0; q < 4; ++q) {
      int ncol0 = wv * 64 + q * 16;
      v8f acc = {0.f, 0.f, 0.f, 0.f, 0.f, 0.f, 0.f, 0.f};
      for (int k = 0; k < 64; k += 4) {
        int kk = k + 2 * half;
        v2f a, b;
        a.x = sT1[lid][kk];     a.y = sT1[lid][kk + 1];
        b.x = w12[kk * FDIM + ncol0 + lid];
        b.y = w12[(kk + 1) * FDIM + ncol0 + lid];
        acc = wmma_f32_k4(a, b, acc);
      }
#pragma unroll
      for (int r = 0; r < 8; ++r) {
        int row = r + 8 * half, col = ncol0 + lid;
        sY[row][col] = acc[r] + b12[col] + sX1[row][col];
      }
    }
  }
  __syncthreads();

  // ---- Phase D: LN2 + write out
  {
    int row = tid >> 3, sub = tid & 7;
    int grow = R0 + row;
    float s = 0.f, s2 = 0.f;
#pragma unroll 8
    for (int j = 0; j < 32; ++j) {
      int f = sub * 32 + j;
      float v = sY[row][f];
      s += v; s2 += v * v;
    }
#pragma unroll
    for (int m = 4; m >= 1; m >>= 1) { s += __shfl_xor(s, m, 32); s2 += __shfl_xor(s2, m, 32); }
    float mean = s * (1.0f / FDIM);
    float var  = s2 * (1.0f / FDIM) - mean * mean;
    float rstd = rsqrtf(var + LN_EPS);
    float* o = out + (size_t)grow * FDIM;
#pragma unroll 8
    for (int j = 0; j < 32; ++j) {
      int f = sub * 32 + j;
      o[f] = (sY[row][f] - mean) * rstd * ln2g[f] + ln2b[f];
    }
  }
}

// ---------------------------------------------------------------- launcher
extern "C" void kernel_launch(void* const* d_in, const int* in_sizes, int n_in,
                              void* d_out, int out_size, void* d_ws, size_t ws_size,
                              hipStream_t stream) {
  const float* x     = (const float*)d_in[0];
  const int*   src   = (const int*)d_in[1];
  const int*   dst   = (const int*)d_in[2];
  // d_in[3] = batch (groups are contiguous blocks of NPG nodes; unused)
  const float* gcn_w = (const float*)d_in[4];
  const float* gcn_b = (const float*)d_in[5];
  const float* bng   = (const float*)d_in[6];
  const float* bnb   = (const float*)d_in[7];
  const float* bnm   = (const float*)d_in[8];
  const float* bnv   = (const float*)d_in[9];
  const float* ln1g  = (const float*)d_in[10];
  const float* ln1b  = (const float*)d_in[11];
  const float* w11   = (const float*)d_in[12];
  const float* b11   = (const float*)d_in[13];
  const float* w12   = (const float*)d_in[14];
  const float* b12   = (const float*)d_in[15];
  const float* ln2g  = (const float*)d_in[16];
  const float* ln2b  = (const float*)d_in[17];
  float* out = (float*)d_out;

  const int N = in_sizes[0] / FDIM;   // 131072
  const int E = in_sizes[1];          // 2097152
  const int G = N / NPGc;             // 256

  float* h      = (float*)d_ws;
  float* degf   = h + N;
  float* dinv   = degf + N;
  float* score  = dinv + N;
  float* pooled = score + N;
  int*   topk   = (int*)(pooled + (size_t)G * 3 * FDIM);

  k_gcn <<<N / 8,            256, 0, stream>>>(x, gcn_w, gcn_b, h, degf, N);
  k_deg <<<(E + 255) / 256,  256, 0, stream>>>(dst, degf, E);
  k_dinv<<<(N + 255) / 256,  256, 0, stream>>>(h, degf, dinv, score, N);
  k_edge<<<(E + 255) / 256,  256, 0, stream>>>(src, dst, h, dinv, score, E);
  k_topk<<<G,                256, 0, stream>>>(score, topk);
  k_pool<<<G,               FDIM, 0, stream>>>(x, topk, pooled);
  k_mlp <<<(G * 3) / 16,     128, 0, stream>>>(pooled, bng, bnb, bnm, bnv,
                                               ln1g, ln1b, w11, b11, w12, b12,
                                               ln2g, ln2b, out);
}